// ReModel_Graph_14594298872139
// MI455X (gfx1250) — compile-verified
//
#include <hip/hip_runtime.h>
#include <math.h>

typedef __attribute__((ext_vector_type(2))) float v2f;
typedef __attribute__((ext_vector_type(8))) float v8f;

#define B_    4
#define L_    1024
#define D_    768
#define HH_   12
#define E_    30
#define M_    60
#define P_    870
#define R_    97
#define BLK_  64
#define NH_   12
#define NN_   (B_ * E_)          // 120 graph nodes
#define NEDGE_ (B_ * P_ + NN_)   // 3480 pair edges + 120 self loops

// ---------------------------------------------------------------------------
// WMMA helpers (V_WMMA_F32_16X16X4_F32, wave32, fp32 in / fp32 acc)
// ---------------------------------------------------------------------------
__device__ __forceinline__ v8f wmma4(v2f a, v2f b, v8f c) {
    return __builtin_amdgcn_wmma_f32_16x16x4_f32(false, a, false, b,
                                                 (short)0, c, false, false);
}

__device__ __forceinline__ void store_tile(float* __restrict__ C, const v8f& acc,
                                           const float* __restrict__ bias,
                                           int m0, int col, int M, int N,
                                           int ldc, int act, int half) {
    if (col < N) {
        float bv = bias ? bias[col] : 0.f;
#pragma unroll
        for (int r = 0; r < 8; ++r) {
            int rowi = m0 + r + 8 * half;
            if (rowi < M) {
                float v = acc[r] + bv;
                if (act == 1) v = tanhf(v);
                C[(long)rowi * ldc + col] = v;
            }
        }
    }
}

// ---------------------------------------------------------------------------
// 1. Recover mention token positions from one-hot mention_map [B,M,L]
// ---------------------------------------------------------------------------
__global__ void k_pos(const float* __restrict__ mmap, int* __restrict__ pos) {
    int idx = blockIdx.x * blockDim.x + threadIdx.x;   // B*M
    if (idx >= B_ * M_) return;
    const float* row = mmap + (long)idx * L_;
    int p = 0;
    for (int l = 0; l < L_; ++l) {
        if (row[l] > 0.5f) { p = l; break; }
    }
    pos[idx] = p;
}

// ---------------------------------------------------------------------------
// 2. entity[b,e,d] = log(exp(ctx[p0,d]) + exp(ctx[p1,d]))  (logsumexp pool)
// ---------------------------------------------------------------------------
__global__ void k_entity(const float* __restrict__ ctx, const int* __restrict__ pos,
                         float* __restrict__ ent) {
    long idx = blockIdx.x * (long)blockDim.x + threadIdx.x;  // B*E*D
    if (idx >= (long)B_ * E_ * D_) return;
    int d  = idx % D_;
    int be = idx / D_;
    int b  = be / E_, e = be % E_;
    int p0 = pos[b * M_ + 2 * e];
    int p1 = pos[b * M_ + 2 * e + 1];
    float c0 = ctx[((long)b * L_ + p0) * D_ + d];
    float c1 = ctx[((long)b * L_ + p1) * D_ + d];
    ent[idx] = logf(expf(c0) + expf(c1));
}

// ---------------------------------------------------------------------------
// 3. ea[b,e,h,m] = (att[b,h,p0,m] + att[b,h,p1,m]) / (2 + 1e-10)
// ---------------------------------------------------------------------------
__global__ void k_ea(const float* __restrict__ att, const int* __restrict__ pos,
                     float* __restrict__ ea) {
    long idx = blockIdx.x * (long)blockDim.x + threadIdx.x;  // B*E*HH*L
    if (idx >= (long)B_ * E_ * HH_ * L_) return;
    int m = idx % L_;  long t = idx / L_;
    int h = t % HH_;   t /= HH_;
    int e = t % E_;    int b = t / E_;
    int p0 = pos[b * M_ + 2 * e];
    int p1 = pos[b * M_ + 2 * e + 1];
    const float* ab = att + ((long)b * HH_ + h) * L_ * L_;
    float v = ab[(long)p0 * L_ + m] + ab[(long)p1 * L_ + m];
    ea[idx] = v * (1.0f / (2.0f + 1e-10f));
}

// ---------------------------------------------------------------------------
// 4. ca[b,p,l] = sum_h ea[h0,h,l]*ea[h1,h,l], then row-normalize (+1e-10)
// ---------------------------------------------------------------------------
__global__ void k_ca(const float* __restrict__ ea, const int* __restrict__ hts,
                     float* __restrict__ ca) {
    int row = blockIdx.x;            // b*P + p
    int b = row / P_;
    int h0 = hts[(long)row * 2 + 0];
    int h1 = hts[(long)row * 2 + 1];
    const float* eah = ea + (long)(b * E_ + h0) * HH_ * L_;
    const float* eat = ea + (long)(b * E_ + h1) * HH_ * L_;
    float* out = ca + (long)row * L_;
    __shared__ float red[256];
    float psum = 0.f;
    for (int l = threadIdx.x; l < L_; l += blockDim.x) {
        float v = 0.f;
#pragma unroll
        for (int h = 0; h < HH_; ++h) v += eah[h * L_ + l] * eat[h * L_ + l];
        out[l] = v;
        psum += v;
    }
    red[threadIdx.x] = psum;
    __syncthreads();
    for (int s = 128; s > 0; s >>= 1) {
        if (threadIdx.x < s) red[threadIdx.x] += red[threadIdx.x + s];
        __syncthreads();
    }
    float inv = 1.0f / (red[0] + 1e-10f);
    for (int l = threadIdx.x; l < L_; l += blockDim.x) out[l] *= inv;
}

// ---------------------------------------------------------------------------
// Generic fp32 WMMA GEMM: C = act(A[M,K] * B[K,N] + bias), row-major.
// One wave32 per 16(M)x64(N) strip: 4 accumulators share one A tile per
// K-step -> 4 back-to-back v_wmma per load-wait.  The 4 tile columns are
// addressed off ONE base register with immediate offsets (+64/+128/+192B).
// Requirements (hold at every call site): K % 4 == 0, N % 64 == 0.
// ---------------------------------------------------------------------------
__global__ __launch_bounds__(32)
void k_wmma_gemm(const float* __restrict__ A, const float* __restrict__ Bm,
                 const float* __restrict__ bias, float* __restrict__ C,
                 int M, int N, int K, int lda, int ldb, int ldc,
                 long sA, long sB, long sC, int act) {
    const int lane = threadIdx.x;
    const int half = lane >> 4;
    const int l15  = lane & 15;
    const int m0 = blockIdx.y << 4;
    const int n0 = blockIdx.x << 6;          // 64-wide N strip
    A  += (long)blockIdx.z * sA;
    Bm += (long)blockIdx.z * sB;
    C  += (long)blockIdx.z * sC;

    int ar = m0 + l15; ar = ar < M ? ar : M - 1;      // clamp: OOB never stored
    const int bc = n0 + l15;                          // N % 64 == 0: no clamp

    const float* ap = A + (long)ar * lda + 2 * half;  // 8B aligned (lda even)
    const float* bp = Bm + (long)(2 * half) * ldb + bc;
    const long ldbl = ldb;

    v8f acc0 = {}, acc1 = {}, acc2 = {}, acc3 = {};
#pragma unroll 2
    for (int k = 0; k < K; k += 4) {
        __builtin_prefetch(bp + 32 * ldbl, 0, 1);   // speculative; OOB dropped
        v2f a = *(const v2f*)ap;
        const float* b0 = bp;                        // row 2*half
        const float* b1 = bp + ldbl;                 // row 2*half + 1
        v2f vb0, vb1, vb2, vb3;
        vb0.x = b0[0];  vb1.x = b0[16]; vb2.x = b0[32]; vb3.x = b0[48];
        vb0.y = b1[0];  vb1.y = b1[16]; vb2.y = b1[32]; vb3.y = b1[48];
        acc0 = wmma4(a, vb0, acc0);
        acc1 = wmma4(a, vb1, acc1);
        acc2 = wmma4(a, vb2, acc2);
        acc3 = wmma4(a, vb3, acc3);
        ap += 4;
        bp += 4 * ldbl;
    }
    store_tile(C, acc0, bias, m0, n0 + l15,      M, N, ldc, act, half);
    store_tile(C, acc1, bias, m0, n0 + 16 + l15, M, N, ldc, act, half);
    store_tile(C, acc2, bias, m0, n0 + 32 + l15, M, N, ldc, act, half);
    store_tile(C, acc3, bias, m0, n0 + 48 + l15, M, N, ldc, act, half);
}

// ---------------------------------------------------------------------------
// 8. f_out[g,c] = f_ni[src,c] + f_nj[dst,c] + f_fij[g,c] + egat_bias[c]
// ---------------------------------------------------------------------------
__global__ void k_edge_combine(const float* __restrict__ f_ni, const float* __restrict__ f_nj,
                               const int* __restrict__ hts, const float* __restrict__ egb,
                               float* __restrict__ fout) {
    long idx = blockIdx.x * (long)blockDim.x + threadIdx.x;  // NEDGE_*D_
    if (idx >= (long)NEDGE_ * D_) return;
    int c = idx % D_;
    long g = idx / D_;
    int src_n, dst_n;
    if (g < (long)B_ * P_) {
        int b = g / P_;
        src_n = b * E_ + hts[g * 2 + 0];
        dst_n = b * E_ + hts[g * 2 + 1];
    } else {
        src_n = dst_n = (int)(g - (long)B_ * P_);
    }
    fout[idx] = f_ni[(long)src_n * D_ + c] + f_nj[(long)dst_n * D_ + c]
              + fout[idx] + egb[c];
}

// ---------------------------------------------------------------------------
// 9. scores[g,h] = sum_j leaky_relu(fout[g,h*64+j], 0.2) * attn_vec[h,j]
// ---------------------------------------------------------------------------
__global__ void k_scores(const float* __restrict__ fout, const float* __restrict__ avec,
                         float* __restrict__ scr) {
    int idx = blockIdx.x * blockDim.x + threadIdx.x;  // NEDGE_*NH_
    if (idx >= NEDGE_ * NH_) return;
    int h = idx % NH_;
    int g = idx / NH_;
    const float* f = fout + (long)g * D_ + h * BLK_;
    const float* a = avec + h * BLK_;
    float s = 0.f;
#pragma unroll 8
    for (int j = 0; j < BLK_; ++j) {
        float x = f[j];
        float e = x >= 0.f ? x : 0.2f * x;
        s += e * a[j];
    }
    scr[idx] = s;
}

// ---------------------------------------------------------------------------
// 10. Per-node edge softmax over in-edges + message aggregation.
//     In-edges of node (b,e): pairs (i,e) i!=e at p = i*29 + (i>e ? e : e-1),
//     plus self loop g = B*P + n.  Deterministic reductions, no atomics.
// ---------------------------------------------------------------------------
__global__ void k_aggregate(const float* __restrict__ scr, const float* __restrict__ nproj,
                            float* __restrict__ nodes) {
    int n = blockIdx.x;            // 0..NN_-1
    int b = n / E_, e = n % E_;
    int tid = threadIdx.x;         // 256
    __shared__ float smax[NH_], ssum[NH_];
    __shared__ float red[256];
    const long gself = (long)B_ * P_ + n;

    for (int h = 0; h < NH_; ++h) {            // per-head max
        float m = scr[gself * NH_ + h];
        for (int i = tid; i < E_; i += blockDim.x) {
            if (i == e) continue;
            int p = i * (E_ - 1) + (i > e ? e : e - 1);
            m = fmaxf(m, scr[(long)(b * P_ + p) * NH_ + h]);
        }
        red[tid] = m; __syncthreads();
        for (int s = 128; s > 0; s >>= 1) {
            if (tid < s) red[tid] = fmaxf(red[tid], red[tid + s]);
            __syncthreads();
        }
        if (tid == 0) smax[h] = red[0];
        __syncthreads();
    }
    for (int h = 0; h < NH_; ++h) {            // per-head exp-sum
        float s = (tid == 0) ? expf(scr[gself * NH_ + h] - smax[h]) : 0.f;
        for (int i = tid; i < E_; i += blockDim.x) {
            if (i == e) continue;
            int p = i * (E_ - 1) + (i > e ? e : e - 1);
            s += expf(scr[(long)(b * P_ + p) * NH_ + h] - smax[h]);
        }
        red[tid] = s; __syncthreads();
        for (int s2 = 128; s2 > 0; s2 >>= 1) {
            if (tid < s2) red[tid] += red[tid + s2];
            __syncthreads();
        }
        if (tid == 0) ssum[h] = red[0];
        __syncthreads();
    }
    for (int c = tid; c < D_; c += blockDim.x) {   // aggregate messages
        int h = c >> 6;
        float inv = 1.0f / ssum[h];
        float acc = nproj[(long)n * D_ + c] * expf(scr[gself * NH_ + h] - smax[h]) * inv;
        for (int i = 0; i < E_; ++i) {
            if (i == e) continue;
            int p = i * (E_ - 1) + (i > e ? e : e - 1);
            float al = expf(scr[(long)(b * P_ + p) * NH_ + h] - smax[h]) * inv;
            acc += nproj[(long)(b * E_ + i) * D_ + c] * al;
        }
        nodes[(long)n * D_ + c] = acc;
    }
}

// ---------------------------------------------------------------------------
// 11. cat[g, 0:768]    = nodes[b*E + hts[g][which]]
//     cat[g, 768:1536] = fout[g]  (edge_ctx = pre-activation f_out incl bias)
// ---------------------------------------------------------------------------
__global__ void k_cat(const float* __restrict__ nodes, const float* __restrict__ fout,
                      const int* __restrict__ hts, int which, float* __restrict__ cat) {
    long idx = blockIdx.x * (long)blockDim.x + threadIdx.x;  // B*P*2D
    if (idx >= (long)B_ * P_ * 2 * D_) return;
    int c = idx % (2 * D_);
    long g = idx / (2 * D_);
    int b = g / P_;
    if (c < D_) {
        int ent = hts[g * 2 + which];
        cat[idx] = nodes[(long)(b * E_ + ent) * D_ + c];
    } else {
        cat[idx] = fout[g * D_ + (c - D_)];
    }
}

// ---------------------------------------------------------------------------
// 13. Fused group-bilinear + relation classifier (bl never materialized):
//     Out[p,r] = sum_q a[p,q>>6] * b[p,(q>>12)<<6 | (q&63)] * W_rel[q,r] + b_rel[r]
//     One wave per 16-pair strip holds ALL 7 R-tiles (7 x v8f accumulators);
//     tiles 0-5 are addressed with immediate offsets off one base register,
//     only tile 6 needs a clamped index (R=97).
// ---------------------------------------------------------------------------
__global__ __launch_bounds__(32)
void k_wmma_bilinear(const float* __restrict__ Ah, const float* __restrict__ Bt,
                     const float* __restrict__ Wrel, const float* __restrict__ brel,
                     float* __restrict__ Out) {
    const int lane = threadIdx.x;
    const int half = lane >> 4;
    const int l15  = lane & 15;
    const int m0 = blockIdx.x << 4;
    const int Mrows = B_ * P_;
    int row = m0 + l15;  row = row < Mrows ? row : Mrows - 1;
    const float* ar = Ah + (long)row * D_;
    const float* br = Bt + (long)row * D_;

    int c6 = 96 + l15; c6 = c6 < R_ ? c6 : R_ - 1;   // only tile 6 clips (R=97)
    const int dc6 = c6 - l15;                         // 96 or 96 - l15

    const int base = 2 * half;
    const float* wp = Wrel + (long)base * R_ + l15;
    v8f t0 = {}, t1 = {}, t2 = {}, t3 = {}, t4 = {}, t5 = {}, t6 = {};
    for (int k = 0; k < NH_ * BLK_ * BLK_; k += 4) {
        __builtin_prefetch(wp + 32 * R_, 0, 1);
        int q0 = k + base, q1 = q0 + 1;
        v2f a;
        a.x = ar[q0 >> 6] * br[((q0 >> 12) << 6) | (q0 & 63)];
        a.y = ar[q1 >> 6] * br[((q1 >> 12) << 6) | (q1 & 63)];
        const float* w0 = wp;
        const float* w1 = wp + R_;
        v2f b;
        b.x = w0[0];   b.y = w1[0];   t0 = wmma4(a, b, t0);
        b.x = w0[16];  b.y = w1[16];  t1 = wmma4(a, b, t1);
        b.x = w0[32];  b.y = w1[32];  t2 = wmma4(a, b, t2);
        b.x = w0[48];  b.y = w1[48];  t3 = wmma4(a, b, t3);
        b.x = w0[64];  b.y = w1[64];  t4 = wmma4(a, b, t4);
        b.x = w0[80];  b.y = w1[80];  t5 = wmma4(a, b, t5);
        b.x = w0[dc6]; b.y = w1[dc6]; t6 = wmma4(a, b, t6);
        wp += 4 * R_;
    }
    store_tile(Out, t0, brel, m0, 0  + l15, Mrows, R_, R_, 0, half);
    store_tile(Out, t1, brel, m0, 16 + l15, Mrows, R_, R_, 0, half);
    store_tile(Out, t2, brel, m0, 32 + l15, Mrows, R_, R_, 0, half);
    store_tile(Out, t3, brel, m0, 48 + l15, Mrows, R_, R_, 0, half);
    store_tile(Out, t4, brel, m0, 64 + l15, Mrows, R_, R_, 0, half);
    store_tile(Out, t5, brel, m0, 80 + l15, Mrows, R_, R_, 0, half);
    store_tile(Out, t6, brel, m0, 96 + l15, Mrows, R_, R_, 0, half);
}

// ---------------------------------------------------------------------------
extern "C" void kernel_launch(void* const* d_in, const int* in_sizes, int n_in,
                              void* d_out, int out_size, void* d_ws, size_t ws_size,
                              hipStream_t stream) {
    (void)in_sizes; (void)n_in; (void)out_size; (void)ws_size;
    const float* ctx    = (const float*)d_in[0];
    const float* att    = (const float*)d_in[1];
    const float* mmap   = (const float*)d_in[2];
    /* d_in[3] entity_map: fixed e -> {2e,2e+1}, unused */
    const int*   hts    = (const int*)  d_in[4];
    const float* W_h    = (const float*)d_in[5];
    const float* b_h    = (const float*)d_in[6];
    const float* W_t    = (const float*)d_in[7];
    const float* b_t    = (const float*)d_in[8];
    const float* W_node = (const float*)d_in[9];
    const float* W_ni   = (const float*)d_in[10];
    const float* W_nj   = (const float*)d_in[11];
    const float* W_fij  = (const float*)d_in[12];
    const float* egb    = (const float*)d_in[13];
    const float* avec   = (const float*)d_in[14];
    const float* W_rel  = (const float*)d_in[15];
    const float* b_rel  = (const float*)d_in[16];
    float* out = (float*)d_out;

    // workspace layout (256B aligned blocks)
    char* ws = (char*)d_ws;
    size_t off = 0;
    auto alloc = [&](size_t bytes) -> char* {
        char* p = ws + off;
        off = (off + bytes + 255) & ~(size_t)255;
        return p;
    };
    int*   pos   = (int*)  alloc((size_t)B_ * M_ * sizeof(int));
    float* ent   = (float*)alloc((size_t)B_ * E_ * D_ * 4);
    float* ea    = (float*)alloc((size_t)B_ * E_ * HH_ * L_ * 4);
    float* ca    = (float*)alloc((size_t)B_ * P_ * L_ * 4);
    float* cinfo = (float*)alloc((size_t)B_ * P_ * D_ * 4);
    float* f_ni  = (float*)alloc((size_t)NN_ * D_ * 4);
    float* f_nj  = (float*)alloc((size_t)NN_ * D_ * 4);
    float* nproj = (float*)alloc((size_t)NN_ * D_ * 4);
    float* fout  = (float*)alloc((size_t)NEDGE_ * D_ * 4);
    float* scr   = (float*)alloc((size_t)NEDGE_ * NH_ * 4);
    float* nodes = (float*)alloc((size_t)NN_ * D_ * 4);
    float* newh  = (float*)alloc((size_t)B_ * P_ * D_ * 4);
    float* newt  = (float*)alloc((size_t)B_ * P_ * D_ * 4);
    // cat [3480,1536] overlays ca (+ start of cinfo): both dead by the time
    // cat is first written (ca dead after cinfo GEMM, cinfo dead after f_fij)
    float* cat   = ca;

    // 1. mention positions
    k_pos<<<dim3((B_ * M_ + 63) / 64), dim3(64), 0, stream>>>(mmap, pos);
    // 2. entity logsumexp pooling
    {
        long n = (long)B_ * E_ * D_;
        k_entity<<<dim3((unsigned)((n + 255) / 256)), dim3(256), 0, stream>>>(ctx, pos, ent);
    }
    // 3. ea gather
    {
        long n = (long)B_ * E_ * HH_ * L_;
        k_ea<<<dim3((unsigned)((n + 255) / 256)), dim3(256), 0, stream>>>(att, pos, ea);
    }
    // 4. ca + normalize
    k_ca<<<dim3(B_ * P_), dim3(256), 0, stream>>>(ea, hts, ca);
    // 5. context_info = ca @ context  (batched GEMM, M=870 N=768 K=1024)
    k_wmma_gemm<<<dim3(D_ / 64, (P_ + 15) / 16, B_), dim3(32), 0, stream>>>(
        ca, ctx, nullptr, cinfo, P_, D_, L_, L_, D_, D_,
        (long)P_ * L_, (long)L_ * D_, (long)P_ * D_, 0);
    // 6. node projections: f_ni, f_nj, nproj = x @ {W_ni, W_nj, W_node}
    k_wmma_gemm<<<dim3(D_ / 64, (NN_ + 15) / 16, 1), dim3(32), 0, stream>>>(
        ent, W_ni, nullptr, f_ni, NN_, D_, D_, D_, D_, D_, 0, 0, 0, 0);
    k_wmma_gemm<<<dim3(D_ / 64, (NN_ + 15) / 16, 1), dim3(32), 0, stream>>>(
        ent, W_nj, nullptr, f_nj, NN_, D_, D_, D_, D_, D_, 0, 0, 0, 0);
    k_wmma_gemm<<<dim3(D_ / 64, (NN_ + 15) / 16, 1), dim3(32), 0, stream>>>(
        ent, W_node, nullptr, nproj, NN_, D_, D_, D_, D_, D_, 0, 0, 0, 0);
    // 7. f_fij = efeat @ W_fij  (pair-edge part from cinfo, self-loop part from x)
    k_wmma_gemm<<<dim3(D_ / 64, (B_ * P_ + 15) / 16, 1), dim3(32), 0, stream>>>(
        cinfo, W_fij, nullptr, fout, B_ * P_, D_, D_, D_, D_, D_, 0, 0, 0, 0);
    k_wmma_gemm<<<dim3(D_ / 64, (NN_ + 15) / 16, 1), dim3(32), 0, stream>>>(
        ent, W_fij, nullptr, fout + (long)B_ * P_ * D_, NN_, D_, D_, D_, D_, D_, 0, 0, 0, 0);
    // 8. f_out combine (in place on fout)
    {
        long n = (long)NEDGE_ * D_;
        k_edge_combine<<<dim3((unsigned)((n + 255) / 256)), dim3(256), 0, stream>>>(
            f_ni, f_nj, hts, egb, fout);
    }
    // 9. attention scores
    k_scores<<<dim3((NEDGE_ * NH_ + 255) / 256), dim3(256), 0, stream>>>(fout, avec, scr);
    // 10. edge softmax + aggregation -> nodes
    k_aggregate<<<dim3(NN_), dim3(256), 0, stream>>>(scr, nproj, nodes);
    // 11. new_h = tanh([nodes[h] ; edge_ctx] @ W_h + b_h)
    {
        long n = (long)B_ * P_ * 2 * D_;
        k_cat<<<dim3((unsigned)((n + 255) / 256)), dim3(256), 0, stream>>>(nodes, fout, hts, 0, cat);
        k_wmma_gemm<<<dim3(D_ / 64, (B_ * P_ + 15) / 16, 1), dim3(32), 0, stream>>>(
            cat, W_h, b_h, newh, B_ * P_, D_, 2 * D_, 2 * D_, D_, D_, 0, 0, 0, 1);
        // 12. new_t
        k_cat<<<dim3((unsigned)((n + 255) / 256)), dim3(256), 0, stream>>>(nodes, fout, hts, 1, cat);
        k_wmma_gemm<<<dim3(D_ / 64, (B_ * P_ + 15) / 16, 1), dim3(32), 0, stream>>>(
            cat, W_t, b_t, newt, B_ * P_, D_, 2 * D_, 2 * D_, D_, D_, 0, 0, 0, 1);
    }
    // 13. fused group-bilinear relation classifier (all 7 R-tiles per wave)
    k_wmma_bilinear<<<dim3((B_ * P_ + 15) / 16), dim3(32), 0, stream>>>(
        newh, newt, W_rel, b_rel, out);
}